// HSTUBlock_52793738003232
// MI455X (gfx1250) — compile-verified
//
#include <hip/hip_runtime.h>
#include <hip/hip_bf16.h>
#include <math.h>

// ---------------------------------------------------------------------------
// HSTU block on MI455X (gfx1250), fp32 via V_WMMA_F32_16X16X4_F32.
// B=4, S=2048, D=128, H=4, hd=32.  All GEMMs run on the WMMA pipe.
// Wave tiles widened to 32x64 (2x4 WMMA tiles) to amortize A-loads.
// ---------------------------------------------------------------------------

typedef float v2f __attribute__((ext_vector_type(2)));
typedef float v8f __attribute__((ext_vector_type(8)));

__device__ __forceinline__ v8f wmma4(v2f a, v2f b, v8f c) {
  // 8-arg pattern: (neg_a, A, neg_b, B, c_mod, C, reuse_a, reuse_b)
  return __builtin_amdgcn_wmma_f32_16x16x4_f32(false, a, false, b, (short)0, c,
                                               false, false);
}

__device__ __forceinline__ float silu_f(float x) {
  return x / (1.0f + __expf(-x));
}

#define BB 4
#define SS 2048
#define DD 128
#define HH 4
#define HD 32
#define MM (BB * SS)  // 8192

// ---------------------------------------------------------------------------
// Kernel 1: fused QKVU projection.
//   [8192,128] x [128, 640]  (cols 0..383 = Wq|Wk|Wv with SiLU, 384..639 = Wu)
// One wave per 32x64 output block (2x4 WMMA tiles); 64-col blocks never
// straddle a weight-segment boundary (segments at 0/128/256/384).
// ---------------------------------------------------------------------------
__global__ __launch_bounds__(32) void proj_kernel(
    const float* __restrict__ X,
    const float* __restrict__ Wq, const float* __restrict__ bq,
    const float* __restrict__ Wk, const float* __restrict__ bk,
    const float* __restrict__ Wv, const float* __restrict__ bv,
    const float* __restrict__ Wu, const float* __restrict__ bu,
    float* __restrict__ Q, float* __restrict__ Ko,
    float* __restrict__ V, float* __restrict__ U) {
  const int lane = threadIdx.x;
  const int half = lane >> 4;
  const int l16  = lane & 15;
  const int m0 = blockIdx.x * 32;
  const int n0 = blockIdx.y * 64;

  const float* Wsel; const float* bsel; float* Osel; int ldb; int ncol; bool act;
  if (n0 < 128)      { Wsel = Wq; bsel = bq; Osel = Q;  ldb = 128; ncol = n0;       act = true;  }
  else if (n0 < 256) { Wsel = Wk; bsel = bk; Osel = Ko; ldb = 128; ncol = n0 - 128; act = true;  }
  else if (n0 < 384) { Wsel = Wv; bsel = bv; Osel = V;  ldb = 128; ncol = n0 - 256; act = true;  }
  else               { Wsel = Wu; bsel = bu; Osel = U;  ldb = 256; ncol = n0 - 384; act = false; }

  v8f acc[2][4] = {};
#pragma unroll 2
  for (int k0 = 0; k0 < 128; k0 += 4) {
    v2f a[2], b[4];
#pragma unroll
    for (int mt = 0; mt < 2; ++mt) {
      const float* ap = X + (m0 + mt * 16 + l16) * DD + k0 + 2 * half;
      a[mt].x = ap[0]; a[mt].y = ap[1];
    }
#pragma unroll
    for (int nt = 0; nt < 4; ++nt) {
      const float* bp = Wsel + (k0 + 2 * half) * ldb + ncol + nt * 16 + l16;
      b[nt].x = bp[0]; b[nt].y = bp[ldb];
    }
#pragma unroll
    for (int mt = 0; mt < 2; ++mt)
#pragma unroll
      for (int nt = 0; nt < 4; ++nt)
        acc[mt][nt] = wmma4(a[mt], b[nt], acc[mt][nt]);
  }

#pragma unroll
  for (int mt = 0; mt < 2; ++mt)
#pragma unroll
    for (int nt = 0; nt < 4; ++nt)
#pragma unroll
      for (int j = 0; j < 8; ++j) {
        const int m = m0 + mt * 16 + half * 8 + j;
        const int n = ncol + nt * 16 + l16;
        float o = acc[mt][nt][j] + bsel[n];
        if (act) o = silu_f(o);
        Osel[m * ldb + n] = o;
      }
}

// ---------------------------------------------------------------------------
// Kernel 2: pos_attn[b] = rel(2048x2048) @ V[b](2048x128),
//   rel[i,j] = pos_w[S-1 + j - i], synthesized from an LDS-staged window.
// One wave per 32x64 output block; per block only pos_w[2016-m0 .. +2078] used.
// ---------------------------------------------------------------------------
__global__ __launch_bounds__(32) void posattn_kernel(
    const float* __restrict__ pos_w, const float* __restrict__ V,
    float* __restrict__ P) {
  __shared__ float lw[2079];
  const int lane = threadIdx.x;
  const int half = lane >> 4;
  const int l16  = lane & 15;
  const int m0 = blockIdx.x * 32;
  const int n0 = blockIdx.y * 64;
  const int b  = blockIdx.z;
  const int base = (SS - 1) - (m0 + 31);  // lowest pos_w index needed
  for (int c = lane; c < 2079; c += 32) lw[c] = pos_w[base + c];
  __syncthreads();

  const float* Vb = V + b * SS * DD;
  v8f acc[2][4] = {};
  for (int k0 = 0; k0 < SS; k0 += 4) {
    v2f a[2], bv[4];
#pragma unroll
    for (int mt = 0; mt < 2; ++mt) {
      const int idx = (k0 + 2 * half) + 31 - (mt * 16 + l16);  // k - mlocal + 31
      a[mt].x = lw[idx];
      a[mt].y = lw[idx + 1];
    }
#pragma unroll
    for (int nt = 0; nt < 4; ++nt) {
      const float* bp = Vb + (k0 + 2 * half) * DD + n0 + nt * 16 + l16;
      bv[nt].x = bp[0]; bv[nt].y = bp[DD];
    }
#pragma unroll
    for (int mt = 0; mt < 2; ++mt)
#pragma unroll
      for (int nt = 0; nt < 4; ++nt)
        acc[mt][nt] = wmma4(a[mt], bv[nt], acc[mt][nt]);
  }

#pragma unroll
  for (int mt = 0; mt < 2; ++mt)
#pragma unroll
    for (int nt = 0; nt < 4; ++nt)
#pragma unroll
      for (int j = 0; j < 8; ++j) {
        const int m = m0 + mt * 16 + half * 8 + j;
        const int n = n0 + nt * 16 + l16;
        P[(b * SS + m) * DD + n] = acc[mt][nt][j];
      }
}

// ---------------------------------------------------------------------------
// Kernel 3: streaming causal attention with squared-masked-SiLU scores.
// No softmax => pure linear accumulation, no rescaling state.
// One wave per (b, h, 32-query tile); 32 keys per outer step. Scores are
// bounced through LDS to convert C-layout -> A-layout for the P@V WMMA.
// ---------------------------------------------------------------------------
__global__ __launch_bounds__(32) void attn_kernel(
    const float* __restrict__ Q, const float* __restrict__ K,
    const float* __restrict__ V, float* __restrict__ O) {
  __shared__ float lp[2][16][32];
  const int lane = threadIdx.x;
  const int half = lane >> 4;
  const int l16  = lane & 15;
  const int i0 = blockIdx.x * 32;
  const int b  = blockIdx.z >> 2;
  const int h  = blockIdx.z & 3;
  const float* Qb = Q + b * SS * DD + h * HD;
  const float* Kb = K + b * SS * DD + h * HD;
  const float* Vb = V + b * SS * DD + h * HD;
  const float scale = 0.17677669529663687f;  // 1/sqrt(32)

  v8f acc[2][2] = {};  // 32 queries x 32 head-dims
  for (int j0 = 0; j0 <= i0; j0 += 32) {
    // ---- S = Q Ktile^T (contraction over hd=32), 32 queries x 32 keys ----
    v8f s[2][2] = {};
#pragma unroll
    for (int k0 = 0; k0 < HD; k0 += 4) {
      v2f a[2], bk[2];
#pragma unroll
      for (int mt = 0; mt < 2; ++mt) {
        const float* ap = Qb + (i0 + mt * 16 + l16) * DD + k0 + 2 * half;
        a[mt].x = ap[0]; a[mt].y = ap[1];
      }
#pragma unroll
      for (int kt = 0; kt < 2; ++kt) {
        const float* bp = Kb + (j0 + kt * 16 + l16) * DD + k0 + 2 * half;
        bk[kt].x = bp[0]; bk[kt].y = bp[1];  // contiguous -> b64
      }
#pragma unroll
      for (int mt = 0; mt < 2; ++mt)
#pragma unroll
        for (int kt = 0; kt < 2; ++kt)
          s[mt][kt] = wmma4(a[mt], bk[kt], s[mt][kt]);
    }
    // ---- mask, square, silu; park in LDS row-major (q x key) ----
#pragma unroll
    for (int mt = 0; mt < 2; ++mt)
#pragma unroll
      for (int kt = 0; kt < 2; ++kt)
#pragma unroll
        for (int j = 0; j < 8; ++j) {
          const int qi = i0 + mt * 16 + half * 8 + j;
          const int ki = j0 + kt * 16 + l16;
          const float sv = s[mt][kt][j] * scale;
          const float sq = (ki <= qi) ? sv * sv : 0.0f;
          lp[mt][half * 8 + j][kt * 16 + l16] = silu_f(sq);
        }
    asm volatile("s_wait_dscnt 0" ::: "memory");
    // ---- O += P @ Vtile (contraction over 32 keys) ----
#pragma unroll
    for (int k0 = 0; k0 < 32; k0 += 4) {
      v2f a[2], bv[2];
#pragma unroll
      for (int mt = 0; mt < 2; ++mt) {
        a[mt].x = lp[mt][l16][k0 + 2 * half];
        a[mt].y = lp[mt][l16][k0 + 2 * half + 1];
      }
#pragma unroll
      for (int nt = 0; nt < 2; ++nt) {
        const float* bp = Vb + (j0 + k0 + 2 * half) * DD + nt * 16 + l16;
        bv[nt].x = bp[0]; bv[nt].y = bp[DD];
      }
#pragma unroll
      for (int mt = 0; mt < 2; ++mt)
#pragma unroll
        for (int nt = 0; nt < 2; ++nt)
          acc[mt][nt] = wmma4(a[mt], bv[nt], acc[mt][nt]);
    }
    asm volatile("s_wait_dscnt 0" ::: "memory");
  }

#pragma unroll
  for (int mt = 0; mt < 2; ++mt)
#pragma unroll
    for (int nt = 0; nt < 2; ++nt)
#pragma unroll
      for (int j = 0; j < 8; ++j) {
        const int m = i0 + mt * 16 + half * 8 + j;
        const int n = h * HD + nt * 16 + l16;
        O[(b * SS + m) * DD + n] = acc[mt][nt][j];
      }
}

// ---------------------------------------------------------------------------
// Kernel 4: per-row LayerNorm over concat([self,pos], 256) then * U -> G.
// One wave32 per row; shuffle-based reductions.
// ---------------------------------------------------------------------------
__global__ __launch_bounds__(256) void ln_mul_kernel(
    const float* __restrict__ SelfA, const float* __restrict__ Pos,
    const float* __restrict__ U, const float* __restrict__ gamma,
    const float* __restrict__ beta, float* __restrict__ G) {
  const int wave = threadIdx.x >> 5;
  const int lane = threadIdx.x & 31;
  const int row  = blockIdx.x * 8 + wave;

  float v[8];
#pragma unroll
  for (int t = 0; t < 8; ++t) {
    const int c = t * 32 + lane;
    v[t] = (c < DD) ? SelfA[row * DD + c] : Pos[row * DD + (c - DD)];
  }
  float s = 0.f;
#pragma unroll
  for (int t = 0; t < 8; ++t) s += v[t];
#pragma unroll
  for (int off = 16; off >= 1; off >>= 1) s += __shfl_xor(s, off, 32);
  const float mu = s * (1.0f / 256.0f);
  float q = 0.f;
#pragma unroll
  for (int t = 0; t < 8; ++t) { const float d = v[t] - mu; q += d * d; }
#pragma unroll
  for (int off = 16; off >= 1; off >>= 1) q += __shfl_xor(q, off, 32);
  const float rstd = rsqrtf(q * (1.0f / 256.0f) + 1e-3f);
#pragma unroll
  for (int t = 0; t < 8; ++t) {
    const int c = t * 32 + lane;
    const float a = (v[t] - mu) * rstd * gamma[c] + beta[c];
    G[row * 256 + c] = U[row * 256 + c] * a;
  }
}

// ---------------------------------------------------------------------------
// Kernel 5: out = relu(G(8192x256) @ Wf(256x128) + bf) + x
// One wave per 32x64 output block (2x4 WMMA tiles).
// ---------------------------------------------------------------------------
__global__ __launch_bounds__(32) void final_kernel(
    const float* __restrict__ G, const float* __restrict__ Wf,
    const float* __restrict__ bf, const float* __restrict__ X,
    float* __restrict__ Out) {
  const int lane = threadIdx.x;
  const int half = lane >> 4;
  const int l16  = lane & 15;
  const int m0 = blockIdx.x * 32;
  const int n0 = blockIdx.y * 64;

  v8f acc[2][4] = {};
#pragma unroll 2
  for (int k0 = 0; k0 < 256; k0 += 4) {
    v2f a[2], b[4];
#pragma unroll
    for (int mt = 0; mt < 2; ++mt) {
      const float* ap = G + (m0 + mt * 16 + l16) * 256 + k0 + 2 * half;
      a[mt].x = ap[0]; a[mt].y = ap[1];
    }
#pragma unroll
    for (int nt = 0; nt < 4; ++nt) {
      const float* bp = Wf + (k0 + 2 * half) * DD + n0 + nt * 16 + l16;
      b[nt].x = bp[0]; b[nt].y = bp[DD];
    }
#pragma unroll
    for (int mt = 0; mt < 2; ++mt)
#pragma unroll
      for (int nt = 0; nt < 4; ++nt)
        acc[mt][nt] = wmma4(a[mt], b[nt], acc[mt][nt]);
  }

#pragma unroll
  for (int mt = 0; mt < 2; ++mt)
#pragma unroll
    for (int nt = 0; nt < 4; ++nt)
#pragma unroll
      for (int j = 0; j < 8; ++j) {
        const int m = m0 + mt * 16 + half * 8 + j;
        const int n = n0 + nt * 16 + l16;
        float o = acc[mt][nt][j] + bf[n];
        o = o > 0.0f ? o : 0.0f;
        Out[m * DD + n] = o + X[m * DD + n];
      }
}

// ---------------------------------------------------------------------------
// Host launcher. Workspace layout (floats):
//   Q[1M] | K[1M] | V[1M] | U[2M] | Pos[1M] | Self[1M]   = 28 MB
//   G (2M floats) reuses the dead Q|K region after attention.
// ---------------------------------------------------------------------------
extern "C" void kernel_launch(void* const* d_in, const int* in_sizes, int n_in,
                              void* d_out, int out_size, void* d_ws,
                              size_t ws_size, hipStream_t stream) {
  (void)in_sizes; (void)n_in; (void)out_size; (void)ws_size;
  const float* x     = (const float*)d_in[0];
  const float* Wq    = (const float*)d_in[1];
  const float* bq    = (const float*)d_in[2];
  const float* Wk    = (const float*)d_in[3];
  const float* bk    = (const float*)d_in[4];
  const float* Wv    = (const float*)d_in[5];
  const float* bv    = (const float*)d_in[6];
  const float* Wu    = (const float*)d_in[7];
  const float* bu    = (const float*)d_in[8];
  const float* posw  = (const float*)d_in[9];
  const float* gamma = (const float*)d_in[10];
  const float* beta  = (const float*)d_in[11];
  const float* Wf    = (const float*)d_in[12];
  const float* bf    = (const float*)d_in[13];
  float* out = (float*)d_out;

  float* ws   = (float*)d_ws;
  float* Q    = ws;                 // 8192*128
  float* Kw   = ws + 1048576;       // 8192*128
  float* V    = ws + 2097152;       // 8192*128
  float* U    = ws + 3145728;       // 8192*256
  float* Pos  = ws + 5242880;       // 8192*128
  float* Self = ws + 6291456;       // 8192*128
  float* G    = ws;                 // 8192*256, reuses Q|K after attention

  proj_kernel<<<dim3(MM / 32, 640 / 64), 32, 0, stream>>>(
      x, Wq, bq, Wk, bk, Wv, bv, Wu, bu, Q, Kw, V, U);
  posattn_kernel<<<dim3(SS / 32, DD / 64, BB), 32, 0, stream>>>(posw, V, Pos);
  attn_kernel<<<dim3(SS / 32, 1, BB * HH), 32, 0, stream>>>(Q, Kw, V, Self);
  ln_mul_kernel<<<dim3(MM / 8), 256, 0, stream>>>(Self, Pos, U, gamma, beta, G);
  final_kernel<<<dim3(MM / 32, DD / 64), 32, 0, stream>>>(G, Wf, bf, x, out);
}